// LI_11708080849230
// MI455X (gfx1250) — compile-verified
//
#include <hip/hip_runtime.h>

// Leaky integrator y[t] = BETA*y[t-1] + x[t] over x:(T, B*D) fp32, time-major.
// Blocked scan: 16-step chunks; Y = L @ X_chunk + carry ⊗ beta^(i+1),
// L[i][j] = beta^(i-j) (lower triangular), computed with 4 chained
// V_WMMA_F32_16X16X4_F32. One wave per 16-channel strip; carry chain T/16.
//
// Channel stride is constexpr so all row offsets fold into the 24-bit
// immediate offset field of global_load/store (saddr + 32-bit voffset form).

typedef __attribute__((ext_vector_type(2))) float v2f;
typedef __attribute__((ext_vector_type(8))) float v8f;

#define LI_BETA 0.95f
#define LI_PF   8                      // prefetch distance in chunks

constexpr unsigned SROW = 32u * 1024u; // B*D channels = row stride (elements)
constexpr unsigned CHUNK = 16u * SROW; // 16 time rows (elements)

__global__ __launch_bounds__(256) void li_wmma_scan(const float* __restrict__ x,
                                                    float* __restrict__ out,
                                                    int T) {
    const int lane = threadIdx.x & 31;
    const int waveId = blockIdx.x * (blockDim.x >> 5) + (threadIdx.x >> 5);
    const unsigned c0 = (unsigned)waveId * 16u;  // first channel of strip

    const int n = lane & 15;   // column within strip / A-matrix row M
    const int h = lane >> 4;   // lane half: K offset (A/B), M offset (C/D)

    // ---- A operands: L[m][k], m=n, k=4s+2h (+1 for .y): beta^(m-k) if m>=k.
    v2f a[4];
#pragma unroll
    for (int s = 0; s < 4; ++s) {
        const int k0 = 4 * s + 2 * h;
        float ax = 0.f, ay = 0.f;
        if (n >= k0)     { float v = 1.f; for (int i = 0; i < n - k0; ++i)     v *= LI_BETA; ax = v; }
        if (n >= k0 + 1) { float v = 1.f; for (int i = 0; i < n - k0 - 1; ++i) v *= LI_BETA; ay = v; }
        a[s].x = ax; a[s].y = ay;
    }

    // ---- carry factors per accumulator VGPR r: beta^(M+1), M = r + 8h
    float cf[8];
    {
        float base = 1.f;
        for (int i = 0; i < 8 * h + 1; ++i) base *= LI_BETA;   // beta^(8h+1)
        float v = base;
#pragma unroll
        for (int r = 0; r < 8; ++r) { cf[r] = v; v *= LI_BETA; }
    }

    // 32-bit element offsets (total tensor = 67M elements < 2^31)
    unsigned offL = c0 + (unsigned)n + (unsigned)(2 * h) * SROW;          // load base (row 2h)
    unsigned offS = c0 + (unsigned)n + (unsigned)(8 * h) * SROW;          // store base (row 8h)
    unsigned offP = c0 + (unsigned)n + (unsigned)(LI_PF * 16 + n) * SROW; // prefetch base

    const int nChunks = T >> 4;
    const int cPF = nChunks - LI_PF;
    float carry = 0.f;

    auto loadChunk = [&](v2f b[4], unsigned off) {
#pragma unroll
        for (int s = 0; s < 4; ++s) {
            b[s].x = x[off + (unsigned)(4 * s) * SROW];      // row 4s+2h
            b[s].y = x[off + (unsigned)(4 * s + 1) * SROW];  // row 4s+2h+1
        }
    };

    auto computeStoreChunk = [&](const v2f b[4]) {
        // P = L @ X_chunk : 4 chained fp32 WMMAs (K = 0..15)
        v8f acc = {};
#pragma unroll
        for (int s = 0; s < 4; ++s) {
            acc = __builtin_amdgcn_wmma_f32_16x16x4_f32(
                /*neg_a=*/false, a[s], /*neg_b=*/false, b[s],
                /*c_mod=*/(short)0, acc, /*reuse_a=*/false, /*reuse_b=*/false);
        }
        // Y[M][n] = P[M][n] + beta^(M+1) * carry[n]
        float o[8];
#pragma unroll
        for (int r = 0; r < 8; ++r) o[r] = acc[r] + cf[r] * carry;

        // new carry = row 15 of Y (upper half, VGPR 7) broadcast to both halves
#if __has_builtin(__builtin_amdgcn_permlanex16)
        {
            int src = __builtin_bit_cast(int, o[7]);
            int sw  = __builtin_amdgcn_permlanex16(src, src, 0x76543210, 0xFEDCBA98,
                                                   false, false);
            carry = h ? o[7] : __builtin_bit_cast(float, sw);
        }
#else
        carry = __shfl(o[7], 16 + n, 32);
#endif
        // store: row offsets r*SROW fold into immediate offsets
#pragma unroll
        for (int r = 0; r < 8; ++r) {
            out[offS + (unsigned)r * SROW] = o[r];
        }
        offS += CHUNK;
    };

    v2f bcur[4], bnext[4];
    loadChunk(bcur, offL); offL += CHUNK;

#pragma unroll 2
    for (int c = 0; c + 1 < nChunks; ++c) {
        loadChunk(bnext, offL); offL += CHUNK;           // next chunk in flight
        if (c < cPF) {                                   // far prefetch -> L2/near
            __builtin_prefetch(&x[offP]);
            offP += CHUNK;
        }
        computeStoreChunk(bcur);
#pragma unroll
        for (int s = 0; s < 4; ++s) bcur[s] = bnext[s];
    }
    computeStoreChunk(bcur);                             // last chunk
}

extern "C" void kernel_launch(void* const* d_in, const int* in_sizes, int n_in,
                              void* d_out, int out_size, void* d_ws, size_t ws_size,
                              hipStream_t stream) {
    (void)n_in; (void)d_ws; (void)ws_size; (void)out_size;
    const float* x = (const float*)d_in[0];
    float* out = (float*)d_out;

    const int S = (int)SROW;               // 32768 channels
    const int T = in_sizes[0] / S;         // 2048

    const int wavesNeeded = S / 16;        // one wave per 16-channel strip
    const int threads = 256;               // 8 waves per block
    const int blocks = wavesNeeded / (threads / 32);

    li_wmma_scan<<<blocks, threads, 0, stream>>>(x, out, T);
}